// VoteAggregationModule_83580063580678
// MI455X (gfx1250) — compile-verified
//
#include <hip/hip_runtime.h>

// ---------------------------------------------------------------------------
// VoteAggregation (PointNet++ SA w/ seed-FPS) for MI455X / gfx1250 (wave32).
//   B=32, N=4096, NP=512, NS=16, C=256, HID=128, RADIUS=0.3
// Pipeline: FPS -> ball query -> [gather x W1] wmma GEMM -> BN stats ->
//           [BN+ReLU x W2] wmma GEMM -> stats -> [BN+ReLU x W3] wmma GEMM ->
//           stats -> BN+ReLU+maxpool(NS).
// GEMMs use V_WMMA_F32_16X16X4_F32 (true fp32 MACs; pipeline is HBM-bound at
// 23.3 TB/s so fp32 WMMA already exceeds the roofline). The K-loop is peeled
// so the xyz rows / zero-pad never appear in the hot path, and B fragments
// are double-buffered so loads for k0+4 overlap the WMMAs for k0.
// ---------------------------------------------------------------------------

#define B_    32
#define N_    4096
#define NP_   512
#define NS_   16
#define C_    256
#define HID_  128
#define NCOLS_ (B_ * NP_ * NS_)          // 262144
#define RAD2_   0.09f
#define INV_RAD_ (1.0f / 0.3f)
#define OUT_FEAT_OFF_ (B_ * NP_ * 3)                       // 49152
#define OUT_SI_OFF_   (OUT_FEAT_OFF_ + B_ * HID_ * NP_)    // 2146304

typedef float v2f __attribute__((ext_vector_type(2)));
typedef float v8f __attribute__((ext_vector_type(8)));

// ---------------------------------------------------------------------------
// Kernel 1: furthest point sample on seed_xyz (one block per batch).
// ---------------------------------------------------------------------------
__global__ __launch_bounds__(256) void fps_kernel(const float* __restrict__ seed,
                                                  const float* __restrict__ xyz,
                                                  int* __restrict__ sinds,
                                                  float* __restrict__ out) {
  __shared__ float sx[N_], sy[N_], sz[N_];
  __shared__ float wval[8];
  __shared__ int widx[8];
  __shared__ int s_sel[NP_];
  __shared__ int s_last;

  const int b = blockIdx.x, t = threadIdx.x;
  const int lane = t & 31, wave = t >> 5;

  const float* sp = seed + (size_t)b * N_ * 3;
  for (int i = t; i < N_; i += 256) {
    sx[i] = sp[3 * i + 0];
    sy[i] = sp[3 * i + 1];
    sz[i] = sp[3 * i + 2];
  }
  if (t == 0) { s_sel[0] = 0; s_last = 0; }
  __syncthreads();

  float dist[16];
#pragma unroll
  for (int j = 0; j < 16; j++) dist[j] = 1e10f;

  for (int it = 1; it < NP_; it++) {
    const int last = s_last;
    const float lx = sx[last], ly = sy[last], lz = sz[last];
    float best = -1.0f;
    int bidx = N_;
#pragma unroll
    for (int j = 0; j < 16; j++) {
      const int i = j * 256 + t;
      const float dx = sx[i] - lx, dy = sy[i] - ly, dz = sz[i] - lz;
      const float d = dx * dx + dy * dy + dz * dz;
      const float nd = fminf(dist[j], d);
      dist[j] = nd;
      if (nd > best || (nd == best && i < bidx)) { best = nd; bidx = i; }
    }
    // wave32 reduction (max value, first index on ties == jnp.argmax)
#pragma unroll
    for (int off = 16; off > 0; off >>= 1) {
      const float ov = __shfl_xor(best, off, 32);
      const int oi = __shfl_xor(bidx, off, 32);
      if (ov > best || (ov == best && oi < bidx)) { best = ov; bidx = oi; }
    }
    if (lane == 0) { wval[wave] = best; widx[wave] = bidx; }
    __syncthreads();
    if (t == 0) {
      float bb = wval[0];
      int bi = widx[0];
      for (int w = 1; w < 8; w++) {
        const float v = wval[w];
        const int ix = widx[w];
        if (v > bb || (v == bb && ix < bi)) { bb = v; bi = ix; }
      }
      s_sel[it] = bi;
      s_last = bi;
    }
    __syncthreads();
  }

  const float* xp = xyz + (size_t)b * N_ * 3;
  float* nx = out + (size_t)b * NP_ * 3;
  float* si = out + OUT_SI_OFF_ + (size_t)b * NP_;
  for (int i = t; i < NP_; i += 256) {
    const int id = s_sel[i];
    nx[3 * i + 0] = xp[3 * id + 0];
    nx[3 * i + 1] = xp[3 * id + 1];
    nx[3 * i + 2] = xp[3 * id + 2];
    sinds[b * NP_ + i] = id;
    si[i] = (float)id;
  }
}

// ---------------------------------------------------------------------------
// Kernel 2: ball query — one wave32 per query point; __ballot-ordered
// first-NS-in-radius selection; pad with first hit.
// ---------------------------------------------------------------------------
__global__ __launch_bounds__(256) void ballquery_kernel(const float* __restrict__ xyz,
                                                        const float* __restrict__ newxyz,
                                                        int* __restrict__ idx) {
  const int gwave = (int)((blockIdx.x * 256 + threadIdx.x) >> 5);
  const int lane = threadIdx.x & 31;
  if (gwave >= B_ * NP_) return;
  const int b = gwave >> 9, p = gwave & (NP_ - 1);

  const float* q = newxyz + ((size_t)b * NP_ + p) * 3;
  const float qx = q[0], qy = q[1], qz = q[2];
  const float* xp = xyz + (size_t)b * N_ * 3;
  int* op = idx + (size_t)gwave * NS_;

  int found = 0;
  int firstHit = -1;
  for (int c0 = 0; c0 < N_ && found < NS_; c0 += 32) {
    const int i = c0 + lane;
    const float dx = xp[3 * i + 0] - qx;
    const float dy = xp[3 * i + 1] - qy;
    const float dz = xp[3 * i + 2] - qz;
    const bool pred = (dx * dx + dy * dy + dz * dz) < RAD2_;
    const unsigned mask = (unsigned)__ballot(pred);
    if (firstHit < 0 && mask) firstHit = c0 + __ffs(mask) - 1;
    const int pos = found + __popc(mask & ((1u << lane) - 1u));
    if (pred && pos < NS_) op[pos] = i;
    found += __popc(mask);
  }
  if (firstHit < 0) firstHit = N_ - 1;  // cannot happen (query is in set), safety
  if (found < NS_ && lane >= found && lane < NS_) op[lane] = firstHit;
}

// ---------------------------------------------------------------------------
// WMMA GEMM: Y(128 x NCOLS) = W(128 x K) * X(K x NCOLS)
// MODE 0: X = fused gather [grouped_xyz/R ; features[:, idx]]  (K=259, pad 260)
// MODE 1: X = relu(Yprev * scale[k] + shift[k])                (K=128)
// Block: 256 thr = 8 waves; wave w -> M strip [16w,16w+16); block covers 64
// columns (4 subtiles of 16). W staged in LDS (row stride padded -> no bank
// conflicts). K loop step 4 -> v_wmma_f32_16x16x4_f32; B double-buffered.
// ---------------------------------------------------------------------------
template <int MODE>
__global__ __launch_bounds__(256) void gemm_kernel(
    const float* __restrict__ W, int K, int Kpad, int kstride,
    const float* __restrict__ Xin,                              // MODE0: features, MODE1: Yprev
    const float* __restrict__ scale, const float* __restrict__ shift,
    const int* __restrict__ idx, const float* __restrict__ xyz,
    const float* __restrict__ newxyz, float* __restrict__ Y) {
  extern __shared__ float lds[];
  float* sc = lds + HID_ * kstride;
  float* sh = sc + Kpad;

  const int t = threadIdx.x;
  for (int i = t; i < HID_ * Kpad; i += 256) {
    const int m = i / Kpad, k = i - m * Kpad;
    lds[m * kstride + k] = (k < K) ? W[m * K + k] : 0.0f;
  }
  if (MODE == 1) {
    for (int i = t; i < Kpad; i += 256) { sc[i] = scale[i]; sh[i] = shift[i]; }
  }
  __syncthreads();

  const int wave = t >> 5, lane = t & 31;
  const int lhalf = lane >> 4, l16 = lane & 15;
  const int m0 = wave * 16;
  const int colbase = blockIdx.x * 64;
  const float* arow = lds + (m0 + l16) * kstride + lhalf * 2;

  v8f acc[4];
#pragma unroll
  for (int s4 = 0; s4 < 4; s4++)
#pragma unroll
    for (int r = 0; r < 8; r++) acc[s4][r] = 0.0f;

  int cols[4];
#pragma unroll
  for (int s4 = 0; s4 < 4; s4++) cols[s4] = colbase + s4 * 16 + l16;

  // MODE0 per-subtile gather state (constant-indexed only)
  float gx[4], gy[4], gz[4];
  const float* fb[4] = {nullptr, nullptr, nullptr, nullptr};
  if (MODE == 0) {
#pragma unroll
    for (int s4 = 0; s4 < 4; s4++) {
      const int col = cols[s4];
      const int bp = col >> 4;  // (b,p) uniform per subtile
      const int bb = bp >> 9, p = bp & (NP_ - 1);
      const int n = idx[col];
      const float* xp = xyz + ((size_t)bb * N_ + n) * 3;
      const float* qp = newxyz + ((size_t)bb * NP_ + p) * 3;
      gx[s4] = (xp[0] - qp[0]) * INV_RAD_;
      gy[s4] = (xp[1] - qp[1]) * INV_RAD_;
      gz[s4] = (xp[2] - qp[2]) * INV_RAD_;
      fb[s4] = Xin + (size_t)bb * C_ * N_ + n;
    }
  }

  // ---- prologue: B fragments for k0 = 0 (xyz rows live only here) ----
  v2f bcur[4];
  if (MODE == 0) {
#pragma unroll
    for (int s4 = 0; s4 < 4; s4++) {
      if (lhalf == 0) { bcur[s4].x = gx[s4]; bcur[s4].y = gy[s4]; }      // k=0,1
      else            { bcur[s4].x = gz[s4]; bcur[s4].y = fb[s4][0]; }   // k=2,3 (c=0)
    }
  } else {
    const int kb = lhalf * 2;
    const float s0 = sc[kb], h0 = sh[kb], s1 = sc[kb + 1], h1 = sh[kb + 1];
#pragma unroll
    for (int s4 = 0; s4 < 4; s4++) {
      const float y0 = Xin[(size_t)kb * NCOLS_ + cols[s4]];
      const float y1 = Xin[(size_t)(kb + 1) * NCOLS_ + cols[s4]];
      bcur[s4].x = fmaxf(fmaf(y0, s0, h0), 0.0f);
      bcur[s4].y = fmaxf(fmaf(y1, s1, h1), 0.0f);
    }
  }

  // ---- main loop: guard-free loads, double-buffered ----
  const int kMain = (MODE == 0) ? (Kpad - 4) : Kpad;  // MODE0: stop before 256
  for (int k0 = 4; k0 < kMain; k0 += 4) {
    v2f bnext[4];
    const int kb = k0 + lhalf * 2;
    if (MODE == 0) {
      const size_t c = (size_t)(kb - 3);  // in [1, 252]: always valid
#pragma unroll
      for (int s4 = 0; s4 < 4; s4++) {
        bnext[s4].x = fb[s4][c * N_];
        bnext[s4].y = fb[s4][(c + 1) * N_];
      }
    } else {
      const float s0 = sc[kb], h0 = sh[kb], s1 = sc[kb + 1], h1 = sh[kb + 1];
#pragma unroll
      for (int s4 = 0; s4 < 4; s4++) {
        const float y0 = Xin[(size_t)kb * NCOLS_ + cols[s4]];
        const float y1 = Xin[(size_t)(kb + 1) * NCOLS_ + cols[s4]];
        bnext[s4].x = fmaxf(fmaf(y0, s0, h0), 0.0f);
        bnext[s4].y = fmaxf(fmaf(y1, s1, h1), 0.0f);
      }
    }
    v2f a;
    a.x = arow[(k0 - 4) * 1];
    a.y = arow[(k0 - 4) + 1];
#pragma unroll
    for (int s4 = 0; s4 < 4; s4++)
      acc[s4] = __builtin_amdgcn_wmma_f32_16x16x4_f32(
          false, a, false, bcur[s4], (short)0, acc[s4], false, false);
#pragma unroll
    for (int s4 = 0; s4 < 4; s4++) bcur[s4] = bnext[s4];
  }

  if (MODE == 0) {
    // ---- epilogue tile k0 = 256: channels 253/254/255 + zero pad (k=259) ----
    v2f bnext[4];
#pragma unroll
    for (int s4 = 0; s4 < 4; s4++) {
      if (lhalf == 0) { bnext[s4].x = fb[s4][(size_t)253 * N_]; bnext[s4].y = fb[s4][(size_t)254 * N_]; }
      else            { bnext[s4].x = fb[s4][(size_t)255 * N_]; bnext[s4].y = 0.0f; }
    }
    v2f a;
    a.x = arow[Kpad - 8];
    a.y = arow[Kpad - 8 + 1];
#pragma unroll
    for (int s4 = 0; s4 < 4; s4++)
      acc[s4] = __builtin_amdgcn_wmma_f32_16x16x4_f32(
          false, a, false, bcur[s4], (short)0, acc[s4], false, false);
#pragma unroll
    for (int s4 = 0; s4 < 4; s4++) bcur[s4] = bnext[s4];
  }
  {
    v2f a;
    a.x = arow[Kpad - 4];
    a.y = arow[Kpad - 4 + 1];
#pragma unroll
    for (int s4 = 0; s4 < 4; s4++)
      acc[s4] = __builtin_amdgcn_wmma_f32_16x16x4_f32(
          false, a, false, bcur[s4], (short)0, acc[s4], false, false);
  }

#pragma unroll
  for (int s4 = 0; s4 < 4; s4++) {
    const int col = cols[s4];
#pragma unroll
    for (int r = 0; r < 8; r++) {
      const int row = m0 + lhalf * 8 + r;
      Y[(size_t)row * NCOLS_ + col] = acc[s4][r];
    }
  }
}

// ---------------------------------------------------------------------------
// Per-channel sum / sum-of-squares for BatchNorm training statistics.
// ---------------------------------------------------------------------------
__global__ __launch_bounds__(256) void stats_kernel(const float* __restrict__ Y,
                                                    float* __restrict__ stats) {
  const int ch = blockIdx.y;
  const int CHUNK = NCOLS_ / 32;  // 8192
  const size_t base = (size_t)ch * NCOLS_ + (size_t)blockIdx.x * CHUNK;
  float s = 0.0f, s2 = 0.0f;
  for (int i = threadIdx.x; i < CHUNK; i += 256) {
    const float v = Y[base + i];
    s += v;
    s2 += v * v;
  }
  __shared__ float rs[256], rs2[256];
  const int t = threadIdx.x;
  rs[t] = s; rs2[t] = s2;
  __syncthreads();
  for (int off = 128; off > 0; off >>= 1) {
    if (t < off) { rs[t] += rs[t + off]; rs2[t] += rs2[t + off]; }
    __syncthreads();
  }
  if (t == 0) {
    atomicAdd(&stats[ch], rs[0]);
    atomicAdd(&stats[HID_ + ch], rs2[0]);
  }
}

__global__ void finalize_kernel(const float* __restrict__ stats,
                                const float* __restrict__ g,
                                const float* __restrict__ be,
                                float* __restrict__ ss) {
  const int c = threadIdx.x;  // 128
  const float mean = stats[c] * (1.0f / (float)NCOLS_);
  const float var = stats[HID_ + c] * (1.0f / (float)NCOLS_) - mean * mean;
  const float scl = g[c] * rsqrtf(var + 1e-5f);
  ss[c] = scl;
  ss[HID_ + c] = be[c] - mean * scl;
}

__global__ void zero_kernel(float* __restrict__ p, int n) {
  for (int i = blockIdx.x * 256 + threadIdx.x; i < n; i += gridDim.x * 256) p[i] = 0.0f;
}

// ---------------------------------------------------------------------------
// Final BN+ReLU+max over NS -> new_features (B,128,NP)
// ---------------------------------------------------------------------------
__global__ __launch_bounds__(256) void maxpool_kernel(const float* __restrict__ Y3,
                                                      const float* __restrict__ ss3,
                                                      float* __restrict__ outFeat) {
  const int gid = blockIdx.x * 256 + threadIdx.x;  // (b,ch,p) flat, 2097152
  if (gid >= B_ * HID_ * NP_) return;
  const int p = gid & (NP_ - 1);
  const int ch = (gid >> 9) & (HID_ - 1);
  const int b = gid >> 16;
  const float scl = ss3[ch], shf = ss3[HID_ + ch];
  const float* row = Y3 + (size_t)ch * NCOLS_ + ((size_t)b * NP_ + p) * NS_;
  const float4* r4 = reinterpret_cast<const float4*>(row);
  float m = -1e30f;
#pragma unroll
  for (int q = 0; q < 4; q++) {
    const float4 v = r4[q];
    m = fmaxf(m, fmaxf(fmaxf(fmaxf(v.x * scl + shf, 0.0f), fmaxf(v.y * scl + shf, 0.0f)),
                       fmaxf(fmaxf(v.z * scl + shf, 0.0f), fmaxf(v.w * scl + shf, 0.0f))));
  }
  outFeat[gid] = m;
}

// ---------------------------------------------------------------------------
extern "C" void kernel_launch(void* const* d_in, const int* in_sizes, int n_in,
                              void* d_out, int out_size, void* d_ws, size_t ws_size,
                              hipStream_t stream) {
  (void)in_sizes; (void)n_in; (void)out_size; (void)ws_size;
  const float* xyz      = (const float*)d_in[0];
  const float* features = (const float*)d_in[1];
  const float* seed     = (const float*)d_in[2];
  const float* w1 = (const float*)d_in[3];
  const float* g1 = (const float*)d_in[4];
  const float* be1 = (const float*)d_in[5];
  const float* w2 = (const float*)d_in[6];
  const float* g2 = (const float*)d_in[7];
  const float* be2 = (const float*)d_in[8];
  const float* w3 = (const float*)d_in[9];
  const float* g3 = (const float*)d_in[10];
  const float* be3 = (const float*)d_in[11];
  float* out = (float*)d_out;

  // workspace layout
  const size_t SZY = (size_t)HID_ * NCOLS_;  // 33.5M floats per activation buf
  float* YA = (float*)d_ws;
  float* YB = YA + SZY;
  int* idx = (int*)(YB + SZY);
  int* sinds = idx + (size_t)B_ * NP_ * NS_;
  float* stats = (float*)(sinds + (size_t)B_ * NP_);  // 3 * 256
  float* ss = stats + 3 * 2 * HID_;                   // 3 * 256 scale|shift

  // allow >64KB dynamic LDS for the GEMM kernels (W1 tile = 130 KB)
  (void)hipFuncSetAttribute(reinterpret_cast<const void*>(&gemm_kernel<0>),
                            hipFuncAttributeMaxDynamicSharedMemorySize, 135168);
  (void)hipFuncSetAttribute(reinterpret_cast<const void*>(&gemm_kernel<1>),
                            hipFuncAttributeMaxDynamicSharedMemorySize, 69632);

  zero_kernel<<<2, 256, 0, stream>>>(stats, 3 * 2 * HID_);
  fps_kernel<<<B_, 256, 0, stream>>>(seed, xyz, sinds, out);
  ballquery_kernel<<<(B_ * NP_ * 32) / 256, 256, 0, stream>>>(xyz, out, idx);

  const int nblk = NCOLS_ / 64;  // 4096
  // Layer 1: fused gather GEMM (K=259 padded to 260; LDS row stride 260)
  gemm_kernel<0><<<nblk, 256, (size_t)HID_ * 260 * 4, stream>>>(
      w1, 3 + C_, 260, 260, features, nullptr, nullptr, idx, xyz, out, YA);
  stats_kernel<<<dim3(32, HID_), 256, 0, stream>>>(YA, stats);
  finalize_kernel<<<1, HID_, 0, stream>>>(stats, g1, be1, ss);

  // Layer 2: BN+ReLU fused into B-load (LDS row stride 132 -> no bank conflicts)
  gemm_kernel<1><<<nblk, 256, ((size_t)HID_ * 132 + 2 * HID_) * 4, stream>>>(
      w2, HID_, HID_, 132, YA, ss, ss + HID_, nullptr, nullptr, nullptr, YB);
  stats_kernel<<<dim3(32, HID_), 256, 0, stream>>>(YB, stats + 256);
  finalize_kernel<<<1, HID_, 0, stream>>>(stats + 256, g2, be2, ss + 256);

  // Layer 3 (reuse YA)
  gemm_kernel<1><<<nblk, 256, ((size_t)HID_ * 132 + 2 * HID_) * 4, stream>>>(
      w3, HID_, HID_, 132, YB, ss + 256, ss + 256 + HID_, nullptr, nullptr, nullptr, YA);
  stats_kernel<<<dim3(32, HID_), 256, 0, stream>>>(YA, stats + 512);
  finalize_kernel<<<1, HID_, 0, stream>>>(stats + 512, g3, be3, ss + 512);

  // BN+ReLU+max over NS
  maxpool_kernel<<<(B_ * HID_ * NP_) / 256, 256, 0, stream>>>(YA, ss + 512, out + OUT_FEAT_OFF_);
}